// MultiMIND_79370995630914
// MI455X (gfx1250) — compile-verified
//
#include <hip/hip_runtime.h>

typedef float v2f __attribute__((ext_vector_type(2)));
typedef float v8f __attribute__((ext_vector_type(8)));

// 6-tap gaussian (offsets -2..+3, sigma=2), normalized: exp(-{4,1,0,1,4,9}/8)/sum
#define WG0 0.14096479f
#define WG1 0.20510289f
#define WG2 0.23241243f
#define WG3 0.20510289f
#define WG4 0.14096479f
#define WG5 0.07545254f

__device__ __forceinline__ int refl(int i) {
  i = i < 0 ? -i : i;
  return i > 255 ? 510 - i : i;
}

// One 16x16 dp tile (gaussian-filtered SSD map) for shift (dy,dx).
// g: LDS gray tile, 29 rows x 32 stride, origin (y0-6, x0-6).
// Horizontal 6-tap pass on VALU builds the B matrix (24x16) in-register;
// vertical 6-tap pass is a banded-A f32 WMMA: D = Wv(16x24) * Tmp(24x16).
// Fully branchless: invalid taps/rows are zeroed via cndmask, LDS reads are
// always in-bounds (shifted row clamped to the loaded 29-row region).
__device__ __forceinline__ v8f conv_tile(const float* g, int y0, int x0,
                                         int dy, int dx, int col, int khalf,
                                         const v2f* aReg) {
  constexpr float WG[6] = {WG0, WG1, WG2, WG3, WG4, WG5};
  v8f acc = {0.f, 0.f, 0.f, 0.f, 0.f, 0.f, 0.f, 0.f};
#pragma unroll
  for (int c = 0; c < 6; ++c) {
    v2f bm;
#pragma unroll
    for (int v = 0; v < 2; ++v) {
      int k = 4 * c + khalf + v;                 // K index (tmp row)
      int gr = y0 + k - 2;                       // global image row
      float rowok = (k <= 20 && (unsigned)gr < 256u) ? 1.f : 0.f;
      int lyc = (k + 4) * 32;
      int lsr = k + 4 + dy;                      // shifted LDS row
      lsr = lsr > 28 ? 28 : lsr;                 // clamp (masked anyway)
      int lys = lsr * 32;
      float tmpv = 0.f;
#pragma unroll
      for (int j = 0; j < 6; ++j) {
        int gx = x0 + col + j - 2;               // SAME conv zero padding
        float wj = ((unsigned)gx < 256u) ? WG[j] : 0.f;
        float d = g[lyc + col + j + 4] - g[lys + col + j + 4 + dx];
        tmpv = fmaf(wj, d * d, tmpv);
      }
      bm[v] = tmpv * rowok;
    }
    acc = __builtin_amdgcn_wmma_f32_16x16x4_f32(
        false, aReg[c], false, bm, (short)0, acc, false, false);
  }
  return acc;
}

__global__ void gray_kernel(const float* __restrict__ in0,
                            const float* __restrict__ in1,
                            float* __restrict__ gray) {
  int t = blockIdx.x * 256 + threadIdx.x;
  if (t >= 524288) return;
  int img = t >> 18;
  int r = t & 262143;
  int b = r >> 16;
  int p = r & 65535;
  const float* in = img ? in1 : in0;
  int base = b * 3 * 65536 + p;
  float m = (in[base] + in[base + 65536] + in[base + 131072]) * (1.f / 3.f);
  gray[t] = (m * 0.5f + 0.5f) * 255.f;
}

__global__ void __launch_bounds__(256)
mind_kernel(const float* __restrict__ gray, float* __restrict__ outL,
            float* __restrict__ outE, float* __restrict__ part) {
  __shared__ float g0[29 * 32];
  __shared__ float g1[29 * 32];
  __shared__ float s0[8][256];
  __shared__ float s1[8][256];
  __shared__ float red[256];

  constexpr float WG[6] = {WG0, WG1, WG2, WG3, WG4, WG5};

  const int tid = threadIdx.x;
  const int lane = tid & 31;
  const int w = tid >> 5;                 // wave id 0..7
  const int col = lane & 15;              // B column / A row / D column
  const int khalf = (lane >> 4) << 1;     // K fold for A/B layouts
  const int rhalf = (lane >> 4) << 3;     // row fold for C/D layout

  const int tile = blockIdx.x & 255;
  const int b = blockIdx.x >> 8;
  const int y0 = (tile >> 4) << 4;
  const int x0 = (tile & 15) << 4;

  const float* gb0 = gray + b * 65536;            // label grayscale
  const float* gb1 = gray + 262144 + b * 65536;   // eus grayscale

  // Banded vertical-weight A operand: identical for every shift of this
  // kernel -> build its 12 per-lane values once, keep in VGPRs.
  v2f aReg[6];
#pragma unroll
  for (int c = 0; c < 6; ++c) {
#pragma unroll
    for (int v = 0; v < 2; ++v) {
      int t = 4 * c + khalf + v - col;    // tap index for output row m=col
      float a = 0.f;
#pragma unroll
      for (int q = 0; q < 6; ++q) a = (t == q) ? WG[q] : a;
      aReg[c][v] = a;
    }
  }

  // reflect-padded 29x29 gray tile per image into LDS (stride 32)
  for (int idx = tid; idx < 29 * 32; idx += 256) {
    int ly = idx >> 5, lx = idx & 31;
    int gy = refl(y0 - 6 + ly);
    int gx = refl(x0 - 6 + lx);
    g0[idx] = gb0[gy * 256 + gx];
    g1[idx] = gb1[gy * 256 + gx];
  }
  __syncthreads();

  // ---- pass 1: 8 near shifts (one per wave) -> vx ----
  const int NEAR_S[8] = {30, 31, 32, 39, 40, 47, 48, 49};
  {
    int s = NEAR_S[w];
    int t = s < 40 ? s : s + 1;
    int dy = t / 9 - 4, dx = t % 9 - 4;
    v8f d0 = conv_tile(g0, y0, x0, dy, dx, col, khalf, aReg);
    v8f d1 = conv_tile(g1, y0, x0, dy, dx, col, khalf, aReg);
#pragma unroll
    for (int v = 0; v < 8; ++v) {
      int row = v + rhalf;
      s0[w][row * 16 + col] = d0[v];
      s1[w][row * 16 + col] = d1[v];
    }
  }
  __syncthreads();
  // fixed-order reduction of the 8 slices -> inverse vx (deterministic);
  // each thread owns element tid, so no extra barrier needed before store.
  {
    float a0 = 0.f, a1 = 0.f;
#pragma unroll
    for (int ww = 0; ww < 8; ++ww) { a0 += s0[ww][tid]; a1 += s1[ww][tid]; }
    s0[0][tid] = 1.0f / (a0 * 0.125f + 1e-6f);
    s1[0][tid] = 1.0f / (a1 * 0.125f + 1e-6f);
  }
  __syncthreads();

  // ---- pass 2: all 80 shifts, 10 per wave ----
  float lossacc = 0.f;
  for (int tt = 0; tt < 10; ++tt) {
    int s = w + (tt << 3);
    int t2 = s < 40 ? s : s + 1;
    int dy = t2 / 9 - 4, dx = t2 % 9 - 4;
    v8f d0 = conv_tile(g0, y0, x0, dy, dx, col, khalf, aReg);
    v8f d1 = conv_tile(g1, y0, x0, dy, dx, col, khalf, aReg);
#pragma unroll
    for (int v = 0; v < 8; ++v) {
      int row = v + rhalf;
      float iv0 = s0[0][row * 16 + col];
      float iv1 = s1[0][row * 16 + col];
      float m0 = __expf(-d0[v] * iv0);
      float m1 = __expf(-d1[v] * iv1);
      int o = (b * 80 + s) * 65536 + (y0 + row) * 256 + (x0 + col);
      outL[o] = m0;
      outE[o] = m1;
      lossacc += fabsf(m0 - m1);
    }
  }

  // deterministic block reduction -> per-block partial
  red[tid] = lossacc;
  __syncthreads();
  for (int st = 128; st > 0; st >>= 1) {
    if (tid < st) red[tid] += red[tid + st];
    __syncthreads();
  }
  if (tid == 0) part[blockIdx.x] = red[0];
}

__global__ void finish_kernel(const float* __restrict__ part,
                              float* __restrict__ out) {
  float s = 0.f;
  for (int i = 0; i < 1024; ++i) s += part[i];
  out[0] = s * (1.0f / 20971520.0f);   // /(4*80*256*256)
}

extern "C" void kernel_launch(void* const* d_in, const int* in_sizes, int n_in,
                              void* d_out, int out_size, void* d_ws,
                              size_t ws_size, hipStream_t stream) {
  (void)in_sizes; (void)n_in; (void)out_size; (void)ws_size;
  const float* label = (const float*)d_in[0];
  const float* eus = (const float*)d_in[1];
  float* out = (float*)d_out;
  float* wsF = (float*)d_ws;
  float* gray = wsF;              // 524288 floats
  float* part = wsF + 524288;     // 1024 floats
  float* outL = out + 1;
  float* outE = out + 1 + 20971520;

  gray_kernel<<<2048, 256, 0, stream>>>(label, eus, gray);
  mind_kernel<<<1024, 256, 0, stream>>>(gray, outL, outE, part);
  finish_kernel<<<1, 1, 0, stream>>>(part, out);
}